// BilateralSliceApply_38233798869768
// MI455X (gfx1250) — compile-verified
//
#include <hip/hip_runtime.h>
#include <cstdint>
#include <cstddef>

// HDRNet bilateral slice + apply, MI455X (gfx1250).
// Memory-bound: ~112MB traffic -> ~5us roofline @ 23.3TB/s.
// CDNA5 paths used: global_load_async_to_lds_b128 (ASYNCcnt) to stage the
// bilateral grid row-slab into LDS, s_wait_asynccnt, ds_load_b128 gathers,
// b128 coalesced global loads/stores for guide/image/out.

namespace {
constexpr int Bn   = 4;
constexpr int Hn   = 1024;
constexpr int Wn   = 1024;
constexpr int GD   = 8;
constexpr int GH   = 16;
constexpr int GW   = 16;
constexpr int NC   = 12;              // (NIN+1)*NOUT
constexpr int CH_STRIDE = GD * GH * GW;   // 2048 floats per grid channel
}

__global__ __launch_bounds__(256)
void bilateral_slice_apply_kernel(const float* __restrict__ grid,
                                  const float* __restrict__ guide,
                                  const float* __restrict__ image,
                                  float* __restrict__ out)
{
    // raw: [c12][z8][ysel2][x16]  = 3072 floats (12KB), filled by async DMA
    // rowG: [z8][x16][c pad16]    = 2048 floats (8KB), y-interpolated coeffs
    __shared__ float raw[NC * GD * 2 * GW];
    __shared__ float rowG[GD * GW * 16];

    const int tid = threadIdx.x;
    const int y   = blockIdx.x;
    const int b   = blockIdx.y;

    // ---- per-row y tent weights (HDRNet convention) ----
    const float gy  = (y + 0.5f) * ((float)GH / (float)Hn) - 0.5f;
    const float fy  = floorf(gy);
    const float wy1 = gy - fy;
    const float wy0 = 1.0f - wy1;
    const int   iy  = (int)fy;
    const int   yi0 = min(max(iy, 0), GH - 1);
    const int   yi1 = min(max(iy + 1, 0), GH - 1);

    // ---- stage the two needed grid y-planes into LDS via async DMA ----
    const float* gbase = grid + (size_t)b * (NC * CH_STRIDE);
    const unsigned rawBase = (unsigned)(size_t)(void*)raw;  // low 32b of generic = LDS offset
#pragma unroll
    for (int it = 0; it < 3; ++it) {
        const int i  = tid + it * 256;        // 768 b128 chunks total
        const int k  = i & 3;                 // x-quad within 16
        const int ys = (i >> 2) & 1;          // which y plane
        const int z  = (i >> 3) & 7;
        const int c  = i >> 6;                // 0..11
        const int ysel = ys ? yi1 : yi0;
        const unsigned goff  = (unsigned)((c * CH_STRIDE + z * (GH * GW) + ysel * GW + k * 4) * 4);
        const unsigned laddr = rawBase + (unsigned)(i * 16);
        asm volatile("global_load_async_to_lds_b128 %0, %1, %2"
                     :: "v"(laddr), "v"(goff), "s"(gbase)
                     : "memory");
    }
    asm volatile("s_wait_asynccnt 0" ::: "memory");
    __syncthreads();

    // ---- y-interpolate once per block: rowG[z][x][c] (c padded to 16) ----
    for (int i = tid; i < GD * GW * NC; i += 256) {
        const int c  = i % NC;
        const int zx = i / NC;
        const int x  = zx & 15;
        const int z  = zx >> 4;
        const float v0 = raw[((c * GD + z) * 2 + 0) * GW + x];
        const float v1 = raw[((c * GD + z) * 2 + 1) * GW + x];
        rowG[((z * GW + x) << 4) + c] = wy0 * v0 + wy1 * v1;
    }
    __syncthreads();

    // ---- 4 pixels per thread, float4 coalesced I/O ----
    const int x0 = tid * 4;
    const size_t rowOff = ((size_t)b * Hn + y) * Wn + x0;
    const float4 g4  = *(const float4*)(guide + rowOff);
    const float4 im0 = *(const float4*)(image + ((size_t)(b * 3 + 0) * Hn + y) * Wn + x0);
    const float4 im1 = *(const float4*)(image + ((size_t)(b * 3 + 1) * Hn + y) * Wn + x0);
    const float4 im2 = *(const float4*)(image + ((size_t)(b * 3 + 2) * Hn + y) * Wn + x0);

    const float gv[4]  = {g4.x, g4.y, g4.z, g4.w};
    const float iv0[4] = {im0.x, im0.y, im0.z, im0.w};
    const float iv1[4] = {im1.x, im1.y, im1.z, im1.w};
    const float iv2[4] = {im2.x, im2.y, im2.z, im2.w};

    float r0[4], r1[4], r2[4];

#pragma unroll
    for (int p = 0; p < 4; ++p) {
        // x tent
        const float gx  = (x0 + p + 0.5f) * ((float)GW / (float)Wn) - 0.5f;
        const float fx  = floorf(gx);
        const float wx1 = gx - fx;
        const float wx0 = 1.0f - wx1;
        const int   ixm = (int)fx;
        const int   xi0 = min(max(ixm, 0), GW - 1);
        const int   xi1 = min(max(ixm + 1, 0), GW - 1);
        // z tent from guide
        const float gz  = gv[p] * (float)GD - 0.5f;
        const float fz  = floorf(gz);
        const float wz1 = gz - fz;
        const float wz0 = 1.0f - wz1;
        const int   izm = (int)fz;
        const int   zi0 = min(max(izm, 0), GD - 1);
        const int   zi1 = min(max(izm + 1, 0), GD - 1);

        const float4* c00 = (const float4*)&rowG[(zi0 * GW + xi0) << 4];
        const float4* c01 = (const float4*)&rowG[(zi0 * GW + xi1) << 4];
        const float4* c10 = (const float4*)&rowG[(zi1 * GW + xi0) << 4];
        const float4* c11 = (const float4*)&rowG[(zi1 * GW + xi1) << 4];

        float res[3];
#pragma unroll
        for (int q = 0; q < 3; ++q) {   // q = output channel; lanes c=4q..4q+3 = (A0,A1,A2,bias)
            const float4 a = c00[q], bq = c01[q], cq = c10[q], dq = c11[q];
            const float ax = wz0 * (wx0 * a.x + wx1 * bq.x) + wz1 * (wx0 * cq.x + wx1 * dq.x);
            const float ay = wz0 * (wx0 * a.y + wx1 * bq.y) + wz1 * (wx0 * cq.y + wx1 * dq.y);
            const float az = wz0 * (wx0 * a.z + wx1 * bq.z) + wz1 * (wx0 * cq.z + wx1 * dq.z);
            const float aw = wz0 * (wx0 * a.w + wx1 * bq.w) + wz1 * (wx0 * cq.w + wx1 * dq.w);
            res[q] = ax * iv0[p] + ay * iv1[p] + az * iv2[p] + aw;
        }
        r0[p] = res[0]; r1[p] = res[1]; r2[p] = res[2];
    }

    float* ob = out + ((size_t)(b * 3) * Hn + y) * Wn + x0;
    *(float4*)(ob)                       = make_float4(r0[0], r0[1], r0[2], r0[3]);
    *(float4*)(ob + (size_t)Hn * Wn)     = make_float4(r1[0], r1[1], r1[2], r1[3]);
    *(float4*)(ob + 2 * (size_t)Hn * Wn) = make_float4(r2[0], r2[1], r2[2], r2[3]);
}

extern "C" void kernel_launch(void* const* d_in, const int* in_sizes, int n_in,
                              void* d_out, int out_size, void* d_ws, size_t ws_size,
                              hipStream_t stream) {
    (void)in_sizes; (void)n_in; (void)out_size; (void)d_ws; (void)ws_size;
    const float* grid  = (const float*)d_in[0];
    const float* guide = (const float*)d_in[1];
    const float* image = (const float*)d_in[2];
    float* out = (float*)d_out;

    dim3 blocks(Hn, Bn);   // one block per (row, batch): 1024 px, 256 thr x 4 px
    bilateral_slice_apply_kernel<<<blocks, 256, 0, stream>>>(grid, guide, image, out);
}